// HGAT_9543417332149
// MI455X (gfx1250) — compile-verified
//
#include <hip/hip_runtime.h>
#include <math.h>

#define B_  8
#define N_  2048
#define E_  64
#define F_  64
#define FC_ 128
#define KP_ 64            // K-panel staged in LDS
#define LP_ 66            // padded panel pitch (even -> 8B-aligned v2f reads)
#define NEGV (-1e9f)

typedef __attribute__((ext_vector_type(2))) float v2f;
typedef __attribute__((ext_vector_type(8))) float v8f;

__device__ __forceinline__ v8f wmma4(v2f a, v2f b, v8f c) {
  // D = A(16x4) * B(4x16) + C, fp32 WMMA
  return __builtin_amdgcn_wmma_f32_16x16x4_f32(false, a, false, b, (short)0, c,
                                               false, false);
}

// ---------------- rdeg[n] = 1 / max(sum_m adj[n,m], 1) ----------------
__global__ void k_rdeg(const float* __restrict__ adj, float* __restrict__ rdeg) {
  __shared__ float red[256];
  int n = blockIdx.x;
  float acc = 0.f;
  for (int m = threadIdx.x; m < N_; m += 256) acc += adj[n * N_ + m];
  red[threadIdx.x] = acc;
  __syncthreads();
  for (int s = 128; s > 0; s >>= 1) {
    if (threadIdx.x < s) red[threadIdx.x] += red[threadIdx.x + s];
    __syncthreads();
  }
  if (threadIdx.x == 0) rdeg[n] = 1.0f / fmaxf(red[0], 1.0f);
}

// ------- h = x@W, hind = x@W_ind (WMMA; W/W_ind staged transposed in LDS) -------
__global__ void k_gemm_h(const float* __restrict__ x, const float* __restrict__ W,
                         const float* __restrict__ Wi, float* __restrict__ h,
                         float* __restrict__ hind) {
  __shared__ float lW[F_ * LP_];    // [col][k]
  __shared__ float lWi[F_ * LP_];
  int wave = threadIdx.x >> 5;
  int lane = threadIdx.x & 31;
  int half = lane >> 4, lr = lane & 15;

  // cooperative transposed stage of both 64x64 weight matrices (once)
  for (int i = threadIdx.x; i < F_ * (F_ / 4); i += 128) {
    int kr = i >> 4;            // k row 0..63
    int c4 = (i & 15) * 4;      // col group
    float4 v = *(const float4*)&W[kr * F_ + c4];
    lW[(c4 + 0) * LP_ + kr] = v.x;
    lW[(c4 + 1) * LP_ + kr] = v.y;
    lW[(c4 + 2) * LP_ + kr] = v.z;
    lW[(c4 + 3) * LP_ + kr] = v.w;
    float4 u = *(const float4*)&Wi[kr * F_ + c4];
    lWi[(c4 + 0) * LP_ + kr] = u.x;
    lWi[(c4 + 1) * LP_ + kr] = u.y;
    lWi[(c4 + 2) * LP_ + kr] = u.z;
    lWi[(c4 + 3) * LP_ + kr] = u.w;
  }
  __syncthreads();

  int R = (blockIdx.x * 4 + wave) * 16;   // row base in flattened [B*N, F]
  v8f aw[4] = {}; v8f ai[4] = {};
  for (int k0 = 0; k0 < F_; k0 += 4) {
    int kk = k0 + half * 2;
    v2f a = *(const v2f*)&x[(R + lr) * F_ + kk];     // global_load_b64
#pragma unroll
    for (int c = 0; c < 4; ++c) {
      v2f bw = *(const v2f*)&lW[(c * 16 + lr) * LP_ + kk];   // ds_load_b64
      v2f bi = *(const v2f*)&lWi[(c * 16 + lr) * LP_ + kk];
      aw[c] = wmma4(a, bw, aw[c]);
      ai[c] = wmma4(a, bi, ai[c]);
    }
  }
#pragma unroll
  for (int c = 0; c < 4; ++c)
#pragma unroll
    for (int r = 0; r < 8; ++r) {
      int row = R + r + half * 8, col = c * 16 + lr;
      h[row * F_ + col]    = aw[c][r];
      hind[row * F_ + col] = ai[c][r];
    }
}

// ---------------- s = leaky_relu(h@a_node, 0.2), hgn = h@a_gn ----------------
__global__ void k_rowdots(const float* __restrict__ h, const float* __restrict__ a_node,
                          const float* __restrict__ a_gn, float* __restrict__ s,
                          float* __restrict__ hgn) {
  int row = blockIdx.x * blockDim.x + threadIdx.x;   // over B*N
  float d1 = 0.f, d2 = 0.f;
  for (int f = 0; f < F_; ++f) {
    float v = h[row * F_ + f];
    d1 += v * a_node[f];
    d2 += v * a_gn[f];
  }
  s[row]   = d1 > 0.f ? d1 : 0.2f * d1;
  hgn[row] = d2;
}

// ---------------- per-(b,e) masked softmax stats: max, sum(exp) ----------------
__global__ void k_edge_stats(const float* __restrict__ H, const float* __restrict__ s,
                             float* __restrict__ mo, float* __restrict__ Zo) {
  __shared__ float red[256];
  int be = blockIdx.x;
  int b = be / E_, e = be % E_;
  float mx = NEGV;
  for (int n = threadIdx.x; n < N_; n += 256)
    if (H[n * E_ + e] > 0.f) mx = fmaxf(mx, s[b * N_ + n]);
  red[threadIdx.x] = mx; __syncthreads();
  for (int st = 128; st > 0; st >>= 1) {
    if (threadIdx.x < st) red[threadIdx.x] = fmaxf(red[threadIdx.x], red[threadIdx.x + st]);
    __syncthreads();
  }
  float m = red[0];
  __syncthreads();
  float sm = 0.f;
  for (int n = threadIdx.x; n < N_; n += 256)
    if (H[n * E_ + e] > 0.f) sm += __expf(s[b * N_ + n] - m);
  red[threadIdx.x] = sm; __syncthreads();
  for (int st = 128; st > 0; st >>= 1) {
    if (threadIdx.x < st) red[threadIdx.x] += red[threadIdx.x + st];
    __syncthreads();
  }
  if (threadIdx.x == 0) { mo[be] = m; Zo[be] = fmaxf(red[0], 1e-30f); }
}

// ---- agg = alpha @ h : alpha panel + h panel staged in LDS, WMMA inner loop ----
__global__ void k_agg(const float* __restrict__ H, const float* __restrict__ s,
                      const float* __restrict__ m, const float* __restrict__ Z,
                      const float* __restrict__ h, float* __restrict__ agg) {
  __shared__ float ldsA[16 * LP_];   // alpha [e-row][k]
  __shared__ float ldsB[F_ * LP_];   // h panel [col][k]
  int wave = threadIdx.x >> 5;       // col tile
  int lane = threadIdx.x & 31;
  int half = lane >> 4, lr = lane & 15;
  int b  = blockIdx.x >> 2;          // grid = B * (E/16)
  int et = blockIdx.x & 3;
  int c  = wave;
  v8f acc = {};
  for (int k0 = 0; k0 < N_; k0 += KP_) {
    __syncthreads();
    // alpha panel: 16 e-rows x 64 k (one exp per element, computed once)
    for (int i = threadIdx.x; i < 16 * KP_; i += 128) {
      int er = i >> 6, k = i & 63;
      int e = et * 16 + er;
      int n = k0 + k;
      float val = 0.f;
      if (H[n * E_ + e] > 0.f)
        val = __expf(s[b * N_ + n] - m[b * E_ + e]) / Z[b * E_ + e];
      ldsA[er * LP_ + k] = val;
    }
    // h panel transposed: [col][k]
    for (int i = threadIdx.x; i < KP_ * (F_ / 4); i += 128) {
      int kr = i >> 4;
      int c4 = (i & 15) * 4;
      float4 v = *(const float4*)&h[(b * N_ + k0 + kr) * F_ + c4];
      ldsB[(c4 + 0) * LP_ + kr] = v.x;
      ldsB[(c4 + 1) * LP_ + kr] = v.y;
      ldsB[(c4 + 2) * LP_ + kr] = v.z;
      ldsB[(c4 + 3) * LP_ + kr] = v.w;
    }
    __syncthreads();
    for (int kq = 0; kq < KP_; kq += 4) {
      int kk = kq + half * 2;
      v2f a  = *(const v2f*)&ldsA[lr * LP_ + kk];
      v2f bh = *(const v2f*)&ldsB[(c * 16 + lr) * LP_ + kk];
      acc = wmma4(a, bh, acc);
    }
  }
#pragma unroll
  for (int r = 0; r < 8; ++r) {
    int row = et * 16 + r + half * 8;
    agg[(b * E_ + row) * F_ + c * 16 + lr] = acc[r];
  }
}

// ---- per-(b,e): se = tanh(elu(agg)@fcW+fcb)@av ; gate = agg@a_ge ----
__global__ void k_edge_feat(const float* __restrict__ agg, const float* __restrict__ fcW,
                            const float* __restrict__ fcb, const float* __restrict__ av,
                            const float* __restrict__ a_ge, float* __restrict__ se,
                            float* __restrict__ gate) {
  __shared__ float red[FC_];
  __shared__ float he[F_];
  int be = blockIdx.x;
  int t = threadIdx.x;               // 128 threads
  if (t < F_) {
    float v = agg[be * F_ + t];
    he[t] = v > 0.f ? v : expm1f(v); // elu
  }
  __syncthreads();
  float acc = fcb[t];
  for (int f = 0; f < F_; ++f) acc += he[f] * fcW[f * FC_ + t];
  red[t] = tanhf(acc) * av[t];
  __syncthreads();
  for (int st = 64; st > 0; st >>= 1) {
    if (t < st) red[t] += red[t + st];
    __syncthreads();
  }
  if (t == 0) se[be] = red[0];
  __syncthreads();
  red[t] = (t < F_) ? agg[be * F_ + t] * a_ge[t] : 0.f;
  __syncthreads();
  for (int st = 64; st > 0; st >>= 1) {
    if (t < st) red[t] += red[t + st];
    __syncthreads();
  }
  if (t == 0) gate[be] = red[0];
}

// ---- industry = diag(rdeg) * (adj @ hind): hind panel staged in LDS (dominant) ----
__global__ void k_industry(const float* __restrict__ adj, const float* __restrict__ hind,
                           const float* __restrict__ rdeg, float* __restrict__ ind) {
  __shared__ float ldsB[F_ * LP_];   // hind panel [col][k]
  int wave = threadIdx.x >> 5;
  int lane = threadIdx.x & 31;
  int half = lane >> 4, lr = lane & 15;
  int grp = blockIdx.x;              // over B * (N/64) = 256 blocks
  int b = grp >> 5;                  // N/64 = 32
  int R = (grp & 31) * 64 + wave * 16;   // wave's 16-row base (node index in batch)
  v8f acc[4] = {};
  for (int k0 = 0; k0 < N_; k0 += KP_) {
    __syncthreads();
    for (int i = threadIdx.x; i < KP_ * (F_ / 4); i += 128) {
      int kr = i >> 4;
      int c4 = (i & 15) * 4;
      float4 v = *(const float4*)&hind[(b * N_ + k0 + kr) * F_ + c4];
      ldsB[(c4 + 0) * LP_ + kr] = v.x;
      ldsB[(c4 + 1) * LP_ + kr] = v.y;
      ldsB[(c4 + 2) * LP_ + kr] = v.z;
      ldsB[(c4 + 3) * LP_ + kr] = v.w;
    }
    __syncthreads();
    for (int kq = 0; kq < KP_; kq += 4) {
      int kk = kq + half * 2;
      v2f a = *(const v2f*)&adj[(R + lr) * N_ + k0 + kk];   // global_load_b64 (L2-resident)
#pragma unroll
      for (int c = 0; c < 4; ++c) {
        v2f bh = *(const v2f*)&ldsB[(c * 16 + lr) * LP_ + kk];  // ds_load_b64
        acc[c] = wmma4(a, bh, acc[c]);
      }
    }
  }
#pragma unroll
  for (int c = 0; c < 4; ++c)
#pragma unroll
    for (int r = 0; r < 8; ++r) {
      int row = R + r + half * 8;
      ind[(b * N_ + row) * F_ + c * 16 + lr] = acc[c][r] * rdeg[row];
    }
}

// ---- fused: coefs1 softmax over E + g = sum_e c1 * elu(beta*h + (1-beta)*agg) ----
__global__ void k_g(const float* __restrict__ H, const float* __restrict__ se,
                    const float* __restrict__ hgn, const float* __restrict__ gate,
                    const float* __restrict__ h, const float* __restrict__ agg,
                    float* __restrict__ g) {
  int wave = threadIdx.x >> 5;
  int lane = threadIdx.x & 31;
  int bn = blockIdx.x * 4 + wave;    // over B*N
  int b = bn >> 11;                  // N = 2048
  int n = bn & (N_ - 1);
  float l0 = H[n * E_ + lane]      > 0.f ? se[b * E_ + lane]      : NEGV;
  float l1 = H[n * E_ + lane + 32] > 0.f ? se[b * E_ + lane + 32] : NEGV;
  float mx = fmaxf(l0, l1);
  for (int d = 16; d > 0; d >>= 1) mx = fmaxf(mx, __shfl_xor(mx, d, 32));
  float ssum = __expf(l0 - mx) + __expf(l1 - mx);
  for (int d = 16; d > 0; d >>= 1) ssum += __shfl_xor(ssum, d, 32);
  float rz = 1.0f / ssum;
  float hg = hgn[b * N_ + n];
  float h0 = h[(b * N_ + n) * F_ + lane];
  float h1 = h[(b * N_ + n) * F_ + lane + 32];
  float g0 = 0.f, g1 = 0.f;
  for (int e = 0; e < E_; ++e) {
    if (H[n * E_ + e] > 0.f) {       // non-members: coef is exactly 0
      float c1 = __expf(se[b * E_ + e] - mx) * rz;
      float beta = 1.0f / (1.0f + __expf(-(hg + gate[b * E_ + e])));
      float a0 = agg[(b * E_ + e) * F_ + lane];
      float a1 = agg[(b * E_ + e) * F_ + lane + 32];
      float v0 = beta * h0 + (1.0f - beta) * a0;
      float v1 = beta * h1 + (1.0f - beta) * a1;
      v0 = v0 > 0.f ? v0 : expm1f(v0);   // elu
      v1 = v1 > 0.f ? v1 : expm1f(v1);
      g0 += c1 * v0; g1 += c1 * v1;
    }
  }
  g[(b * N_ + n) * F_ + lane]      = g0;
  g[(b * N_ + n) * F_ + lane + 32] = g1;
}

// ---- final: 2-way attention over [industry, g] per (b,n) ----
__global__ void k_final(const float* __restrict__ ind, const float* __restrict__ g,
                        const float* __restrict__ fcW, const float* __restrict__ fcb,
                        const float* __restrict__ av, float* __restrict__ out) {
  __shared__ float lds[4][2][F_];
  int wave = threadIdx.x >> 5;
  int lane = threadIdx.x & 31;
  int bn = blockIdx.x * 4 + wave;    // over B*N
  const float* indp = ind + bn * F_;
  const float* gp   = g   + bn * F_;
  lds[wave][0][lane]      = indp[lane];
  lds[wave][0][lane + 32] = indp[lane + 32];
  lds[wave][1][lane]      = gp[lane];
  lds[wave][1][lane + 32] = gp[lane + 32];
  __syncthreads();
  float si = 0.f, sg = 0.f;
#pragma unroll
  for (int j = 0; j < 4; ++j) {
    int col = lane + 32 * j;
    float ai = fcb[col], ag = fcb[col];
    for (int f = 0; f < F_; ++f) {
      float w = fcW[f * FC_ + col];
      ai += lds[wave][0][f] * w;
      ag += lds[wave][1][f] * w;
    }
    float a = av[col];
    si += tanhf(ai) * a;
    sg += tanhf(ag) * a;
  }
  for (int d = 16; d > 0; d >>= 1) {
    si += __shfl_xor(si, d, 32);
    sg += __shfl_xor(sg, d, 32);
  }
  float mx = fmaxf(si, sg);
  float wi = __expf(si - mx), wg = __expf(sg - mx);
  float rs = 1.0f / (wi + wg);
  wi *= rs; wg *= rs;
  out[bn * F_ + lane]      = wi * lds[wave][0][lane]      + wg * lds[wave][1][lane];
  out[bn * F_ + lane + 32] = wi * lds[wave][0][lane + 32] + wg * lds[wave][1][lane + 32];
}

extern "C" void kernel_launch(void* const* d_in, const int* in_sizes, int n_in,
                              void* d_out, int out_size, void* d_ws, size_t ws_size,
                              hipStream_t stream) {
  (void)in_sizes; (void)n_in; (void)out_size; (void)ws_size;
  const float* x      = (const float*)d_in[0];
  const float* H      = (const float*)d_in[1];
  const float* adj    = (const float*)d_in[2];
  // d_in[3] = nhid (constant 64, baked in)
  const float* W      = (const float*)d_in[4];
  const float* Wi     = (const float*)d_in[5];
  const float* a_node = (const float*)d_in[6];
  const float* a_gn   = (const float*)d_in[7];
  const float* a_ge   = (const float*)d_in[8];
  const float* fcW    = (const float*)d_in[9];
  const float* fcb    = (const float*)d_in[10];
  const float* av     = (const float*)d_in[11];
  float* out = (float*)d_out;

  float* ws   = (float*)d_ws;
  float* h    = ws;                 // B*N*F  = 1048576
  float* hind = h    + 1048576;     // 1048576
  float* s    = hind + 1048576;     // B*N    = 16384
  float* hgn  = s    + 16384;       // 16384
  float* rdeg = hgn  + 16384;       // 2048
  float* m    = rdeg + 2048;        // B*E = 512
  float* Z    = m    + 512;         // 512
  float* agg  = Z    + 512;         // B*E*F  = 32768
  float* gate = agg  + 32768;       // 512
  float* se   = gate + 512;         // 512
  float* ind  = se   + 512;         // 1048576
  float* g    = ind  + 1048576;     // 1048576   (total ~16.3 MB)

  k_rdeg      <<<N_, 256, 0, stream>>>(adj, rdeg);
  k_gemm_h    <<<(B_ * N_ / 16) / 4, 128, 0, stream>>>(x, W, Wi, h, hind);
  k_rowdots   <<<(B_ * N_) / 256, 256, 0, stream>>>(h, a_node, a_gn, s, hgn);
  k_edge_stats<<<B_ * E_, 256, 0, stream>>>(H, s, m, Z);
  k_agg       <<<B_ * (E_ / 16), 128, 0, stream>>>(H, s, m, Z, h, agg);
  k_edge_feat <<<B_ * E_, 128, 0, stream>>>(agg, fcW, fcb, av, a_ge, se, gate);
  k_industry  <<<B_ * (N_ / 64), 128, 0, stream>>>(adj, hind, rdeg, ind);
  k_g         <<<(B_ * N_) / 4, 128, 0, stream>>>(H, se, hgn, gate, h, agg, g);
  k_final     <<<(B_ * N_) / 4, 128, 0, stream>>>(ind, g, fcW, fcb, av, out);
}